// GCN2_79946521247965
// MI455X (gfx1250) — compile-verified
//
#include <hip/hip_runtime.h>
#include <math.h>

#define N_NODES  100000
#define N_EDGES  3200000
#define IN_DIM   128
#define HID      64
#define OUT_DIM  10
#define N_GRAPHS 64

typedef __attribute__((ext_vector_type(2))) float v2f;
typedef __attribute__((ext_vector_type(8))) float v8f;

// ---------------------------------------------------------------------------
// Dense linear: H[node, 0:64] = X[node, 0:K] @ W[64, K]^T + b
// One wave32 computes one 16x16 output tile via V_WMMA_F32_16X16X4_F32.
// A (16x4 f32): lane l -> row M=l&15, holds K = (l>>4)*2 + {0,1}
// B (4x16 f32): lane l -> col N=l&15, holds K = (l>>4)*2 + {0,1}
// D (16x16 f32): VGPR j -> M = j + 8*(l>>4), N = l&15
// ---------------------------------------------------------------------------
__global__ __launch_bounds__(128)
void gcn_linear_wmma(const float* __restrict__ X, const float* __restrict__ W,
                     const float* __restrict__ b, float* __restrict__ H, int K) {
    const int lane  = threadIdx.x & 31;
    const int wave  = threadIdx.x >> 5;
    const int tile  = blockIdx.x * 4 + wave;   // (N_NODES/16) * (HID/16) tiles
    const int mtile = tile >> 2;               // node tile
    const int ntile = tile & 3;                // output-feature tile (0..3)
    const int row0  = mtile * 16;
    const int col0  = ntile * 16;

    const int mn    = lane & 15;
    const int khalf = (lane >> 4) * 2;         // 0 or 2

    const float* __restrict__ Arow = X + (size_t)(row0 + mn) * K + khalf;
    const float* __restrict__ Brow = W + (size_t)(col0 + mn) * K + khalf;

    v8f c = {};
    for (int k0 = 0; k0 < K; k0 += 4) {
        v2f a  = *(const v2f*)(Arow + k0);
        v2f bb = *(const v2f*)(Brow + k0);
        c = __builtin_amdgcn_wmma_f32_16x16x4_f32(
                /*neg_a=*/false, a, /*neg_b=*/false, bb,
                /*c_mod=*/(short)0, c, /*reuse_a=*/false, /*reuse_b=*/false);
    }

    const float bias = b[col0 + mn];
    float* outp = H + (size_t)(row0 + 8 * (lane >> 4)) * HID + col0 + mn;
#pragma unroll
    for (int j = 0; j < 8; ++j)
        outp[(size_t)j * HID] = c[j] + bias;
}

// ---------------------------------------------------------------------------
// SpMM scatter: Hout[rows[e], :] += vals[e] * Hin[cols[e], :]
// One wave per edge; each lane handles a contiguous float2 of features.
// Hin (25.6 MB) fits in L2, so random gathers are L2 hits; scatter uses the
// hardware global_atomic_add_f32 path.
// ---------------------------------------------------------------------------
__global__ __launch_bounds__(256)
void spmm_scatter(const int* __restrict__ rows, const int* __restrict__ cols,
                  const float* __restrict__ vals,
                  const float* __restrict__ Hin, float* __restrict__ Hout) {
    const int lane = threadIdx.x & 31;
    const int edge = blockIdx.x * 8 + (threadIdx.x >> 5);
    if (edge >= N_EDGES) return;

    const int   r = rows[edge];
    const int   s = cols[edge];
    const float v = vals[edge];

    const float2 h = *(const float2*)(Hin + (size_t)s * HID + lane * 2);
    float* dst = Hout + (size_t)r * HID + lane * 2;
    unsafeAtomicAdd(dst + 0, v * h.x);
    unsafeAtomicAdd(dst + 1, v * h.y);
}

// ReLU in place; Sum = accumulate ? Sum + relu : relu
__global__ __launch_bounds__(256)
void relu_accum(float* __restrict__ Xc, float* __restrict__ Xs,
                int accumulate, size_t n) {
    size_t i = (size_t)blockIdx.x * blockDim.x + threadIdx.x;
    if (i >= n) return;
    float r = fmaxf(Xc[i], 0.0f);
    Xc[i] = r;
    Xs[i] = accumulate ? (Xs[i] + r) : r;
}

// Segment-sum pooling: sums[batch[node], :] += Sum[node, :]/3 ; counts += 1
__global__ __launch_bounds__(256)
void pool_scatter(const float* __restrict__ Xs, const int* __restrict__ batch,
                  float* __restrict__ sums, float* __restrict__ counts) {
    const int lane = threadIdx.x & 31;
    const int node = blockIdx.x * 8 + (threadIdx.x >> 5);
    if (node >= N_NODES) return;
    const int g = batch[node];
    const float2 h = *(const float2*)(Xs + (size_t)node * HID + lane * 2);
    float* dst = sums + (size_t)g * HID + lane * 2;
    unsafeAtomicAdd(dst + 0, h.x * (1.0f / 3.0f));
    unsafeAtomicAdd(dst + 1, h.y * (1.0f / 3.0f));
    if (lane == 0) unsafeAtomicAdd(counts + g, 1.0f);
}

// Head: pooled = sums/max(counts,1); logits = pooled @ Wout^T + bout; softmax.
// One thread per graph (64 threads, trivial work).
__global__ __launch_bounds__(64)
void head_kernel(const float* __restrict__ sums, const float* __restrict__ counts,
                 const float* __restrict__ Wout, const float* __restrict__ bout,
                 float* __restrict__ out) {
    const int g = threadIdx.x;
    if (g >= N_GRAPHS) return;
    const float inv_cnt = 1.0f / fmaxf(counts[g], 1.0f);

    float pooled[HID];
#pragma unroll
    for (int j = 0; j < HID; ++j)
        pooled[j] = sums[(size_t)g * HID + j] * inv_cnt;

    float logits[OUT_DIM];
    float m = -INFINITY;
    for (int o = 0; o < OUT_DIM; ++o) {
        float acc = bout[o];
#pragma unroll
        for (int j = 0; j < HID; ++j)
            acc += pooled[j] * Wout[(size_t)o * HID + j];
        logits[o] = acc;
        m = fmaxf(m, acc);
    }
    float se = 0.0f;
    for (int o = 0; o < OUT_DIM; ++o) {
        logits[o] = __expf(logits[o] - m);
        se += logits[o];
    }
    const float inv = 1.0f / se;
    for (int o = 0; o < OUT_DIM; ++o)
        out[(size_t)g * OUT_DIM + o] = logits[o] * inv;
}

// ---------------------------------------------------------------------------
extern "C" void kernel_launch(void* const* d_in, const int* in_sizes, int n_in,
                              void* d_out, int out_size, void* d_ws, size_t ws_size,
                              hipStream_t stream) {
    const float* X    = (const float*)d_in[0];
    const float* vals = (const float*)d_in[1];
    const float* W1   = (const float*)d_in[2];
    const float* b1   = (const float*)d_in[3];
    const float* W2   = (const float*)d_in[4];
    const float* b2   = (const float*)d_in[5];
    const float* W3   = (const float*)d_in[6];
    const float* b3   = (const float*)d_in[7];
    const float* Wout = (const float*)d_in[8];
    const float* bout = (const float*)d_in[9];
    const int*   rows = (const int*)d_in[10];
    const int*   cols = (const int*)d_in[11];
    const int*   batc = (const int*)d_in[12];
    float*       out  = (float*)d_out;

    const size_t NH = (size_t)N_NODES * HID;
    float* H      = (float*)d_ws;          // linear output        (25.6 MB)
    float* Cur    = H   + NH;              // current layer X_l    (25.6 MB)
    float* Sum    = Cur + NH;              // X1+X2+X3 accumulator (25.6 MB)
    float* sums   = Sum + NH;              // pooled sums (64x64)
    float* counts = sums + (size_t)N_GRAPHS * HID;

    const int lin_blocks  = (N_NODES / 16) * (HID / 16) / 4;   // 6250
    const int edge_blocks = N_EDGES / 8;                       // 400000
    const int elem_blocks = (int)((NH + 255) / 256);           // 25000
    const int pool_blocks = (N_NODES + 7) / 8;                 // 12500

    // ---- Layer 1: K = 128
    gcn_linear_wmma<<<lin_blocks, 128, 0, stream>>>(X, W1, b1, H, IN_DIM);
    hipMemsetAsync(Cur, 0, NH * sizeof(float), stream);
    spmm_scatter<<<edge_blocks, 256, 0, stream>>>(rows, cols, vals, H, Cur);
    relu_accum<<<elem_blocks, 256, 0, stream>>>(Cur, Sum, 0, NH);

    // ---- Layer 2: K = 64
    gcn_linear_wmma<<<lin_blocks, 128, 0, stream>>>(Cur, W2, b2, H, HID);
    hipMemsetAsync(Cur, 0, NH * sizeof(float), stream);
    spmm_scatter<<<edge_blocks, 256, 0, stream>>>(rows, cols, vals, H, Cur);
    relu_accum<<<elem_blocks, 256, 0, stream>>>(Cur, Sum, 1, NH);

    // ---- Layer 3: K = 64
    gcn_linear_wmma<<<lin_blocks, 128, 0, stream>>>(Cur, W3, b3, H, HID);
    hipMemsetAsync(Cur, 0, NH * sizeof(float), stream);
    spmm_scatter<<<edge_blocks, 256, 0, stream>>>(rows, cols, vals, H, Cur);
    relu_accum<<<elem_blocks, 256, 0, stream>>>(Cur, Sum, 1, NH);

    // ---- Pool + head
    hipMemsetAsync(sums, 0, ((size_t)N_GRAPHS * HID + N_GRAPHS) * sizeof(float), stream);
    pool_scatter<<<pool_blocks, 256, 0, stream>>>(Sum, batc, sums, counts);
    head_kernel<<<1, 64, 0, stream>>>(sums, counts, Wout, bout, out);
}